// AlignmentModule_25262997635486
// MI455X (gfx1250) — compile-verified
//
#include <hip/hip_runtime.h>
#include <hip/hip_bf16.h>
#include <math.h>

// ---------------------------------------------------------------------------
// CDNA5 (gfx1250) fp32 WMMA helpers.  V_WMMA_F32_16X16X4_F32:
//   A: 16x4 f32, 2 VGPRs/lane:  lane<16 -> M=lane, K={0,1}; lane>=16 -> K={2,3}
//   B: 4x16 f32, 2 VGPRs/lane:  lane%16 = N, VGPR v holds K=v (lanes 0-15),
//                               K=v+2 (lanes 16-31)   (mirrors C/D half split)
//   C/D: 16x16 f32, 8 VGPRs:    element e -> M = e + 8*(lane>=16), N = lane%16
// ---------------------------------------------------------------------------
typedef float v2f __attribute__((ext_vector_type(2)));
typedef float v8f __attribute__((ext_vector_type(8)));

__device__ __forceinline__ v8f wmma_f32(v2f a, v2f b, v8f c) {
    return __builtin_amdgcn_wmma_f32_16x16x4_f32(
        /*neg_a=*/false, a, /*neg_b=*/false, b,
        /*c_mod=*/(short)0, c, /*reuse_a=*/false, /*reuse_b=*/false);
}

#define ZERO8 {0.f,0.f,0.f,0.f,0.f,0.f,0.f,0.f}

// LDS row stride (floats) for 16x512 tiles: 516 -> m*516 mod 64 = 4m, bank-conflict free
#define LDST 516

// ===========================================================================
// Unit MLP: out = l2norm( gelu( LN( X@W1 + b1 ) ) @ W2 + b2 )
// X: [R,512], W1/W2: [512,512].  One block = 16 rows x 512 cols, 8 waves.
// ===========================================================================
__global__ __launch_bounds__(256) void unit_mlp_kernel(
    const float* __restrict__ X,
    const float* __restrict__ W1, const float* __restrict__ b1,
    const float* __restrict__ g,  const float* __restrict__ bt,
    const float* __restrict__ W2, const float* __restrict__ b2,
    float* __restrict__ out, int R)
{
    __shared__ float sm[16 * LDST];
    const int tid  = threadIdx.x;
    const int wave = tid >> 5;
    const int lane = tid & 31;
    const int half = lane >> 4;
    const int lmod = lane & 15;
    const int row0 = blockIdx.x * 16;
    const int colbase = wave * 64;

    // ---- stage X tile (16x512) into LDS, float4 vectorized ----
    for (int i = tid; i < 16 * 128; i += 256) {
        const int r = i >> 7, c4 = i & 127;
        const float4 v = ((const float4*)(X + (size_t)(row0 + r) * 512))[c4];
        *(float4*)&sm[r * LDST + c4 * 4] = v;
    }
    __syncthreads();

    // ---- layer 1 GEMM: wave computes 16 x 64 (4 WMMA tiles) ----
    v8f acc0 = ZERO8, acc1 = ZERO8, acc2 = ZERO8, acc3 = ZERO8;
    for (int k0 = 0; k0 < 512; k0 += 4) {
        const int ka = k0 + 2 * half;
        v2f a; a.x = sm[lmod * LDST + ka]; a.y = sm[lmod * LDST + ka + 1];
        const float* w0 = W1 + (size_t)ka * 512;
        v2f b;
        b.x = w0[colbase + lmod];        b.y = w0[512 + colbase + lmod];
        acc0 = wmma_f32(a, b, acc0);
        b.x = w0[colbase + 16 + lmod];   b.y = w0[512 + colbase + 16 + lmod];
        acc1 = wmma_f32(a, b, acc1);
        b.x = w0[colbase + 32 + lmod];   b.y = w0[512 + colbase + 32 + lmod];
        acc2 = wmma_f32(a, b, acc2);
        b.x = w0[colbase + 48 + lmod];   b.y = w0[512 + colbase + 48 + lmod];
        acc3 = wmma_f32(a, b, acc3);
    }
    __syncthreads();   // done reading X tile
    // ---- write h = acc + b1 into LDS ----
    {
        v8f* accs[4] = {&acc0, &acc1, &acc2, &acc3};
#pragma unroll
        for (int t = 0; t < 4; ++t) {
            const int col = colbase + t * 16 + lmod;
            const float bias = b1[col];
#pragma unroll
            for (int e = 0; e < 8; ++e)
                sm[(e + 8 * half) * LDST + col] = (*accs[t])[e] + bias;
        }
    }
    __syncthreads();

    // ---- LayerNorm + exact GELU, wave w owns rows {w, w+8} ----
    for (int r = wave; r < 16; r += 8) {
        float s1 = 0.f, s2 = 0.f;
        for (int c = lane; c < 512; c += 32) {
            const float x = sm[r * LDST + c]; s1 += x; s2 += x * x;
        }
#pragma unroll
        for (int off = 16; off > 0; off >>= 1) {
            s1 += __shfl_xor(s1, off); s2 += __shfl_xor(s2, off);
        }
        const float mu   = s1 * (1.0f / 512.0f);
        const float var  = s2 * (1.0f / 512.0f) - mu * mu;
        const float rstd = rsqrtf(var + 1e-5f);
        for (int c = lane; c < 512; c += 32) {
            const float x = (sm[r * LDST + c] - mu) * rstd * g[c] + bt[c];
            sm[r * LDST + c] = 0.5f * x * (1.0f + erff(x * 0.70710678118654752f));
        }
    }
    __syncthreads();

    // ---- layer 2 GEMM ----
    acc0 = (v8f)ZERO8; acc1 = (v8f)ZERO8; acc2 = (v8f)ZERO8; acc3 = (v8f)ZERO8;
    for (int k0 = 0; k0 < 512; k0 += 4) {
        const int ka = k0 + 2 * half;
        v2f a; a.x = sm[lmod * LDST + ka]; a.y = sm[lmod * LDST + ka + 1];
        const float* w0 = W2 + (size_t)ka * 512;
        v2f b;
        b.x = w0[colbase + lmod];        b.y = w0[512 + colbase + lmod];
        acc0 = wmma_f32(a, b, acc0);
        b.x = w0[colbase + 16 + lmod];   b.y = w0[512 + colbase + 16 + lmod];
        acc1 = wmma_f32(a, b, acc1);
        b.x = w0[colbase + 32 + lmod];   b.y = w0[512 + colbase + 32 + lmod];
        acc2 = wmma_f32(a, b, acc2);
        b.x = w0[colbase + 48 + lmod];   b.y = w0[512 + colbase + 48 + lmod];
        acc3 = wmma_f32(a, b, acc3);
    }
    __syncthreads();   // done reading h
    {
        v8f* accs[4] = {&acc0, &acc1, &acc2, &acc3};
#pragma unroll
        for (int t = 0; t < 4; ++t) {
            const int col = colbase + t * 16 + lmod;
            const float bias = b2[col];
#pragma unroll
            for (int e = 0; e < 8; ++e)
                sm[(e + 8 * half) * LDST + col] = (*accs[t])[e] + bias;
        }
    }
    __syncthreads();

    // ---- row L2 norm + store ----
    for (int r = wave; r < 16; r += 8) {
        float ss = 0.f;
        for (int c = lane; c < 512; c += 32) {
            const float x = sm[r * LDST + c]; ss += x * x;
        }
#pragma unroll
        for (int off = 16; off > 0; off >>= 1) ss += __shfl_xor(ss, off);
        const float inv = 1.0f / fmaxf(sqrtf(ss), 1e-12f);
        float* o = out + (size_t)(row0 + r) * 512;
        for (int c = lane; c < 512; c += 32) o[c] = sm[r * LDST + c] * inv;
    }
}

// ===========================================================================
// Global projection: out = l2norm( LN( X@W + b ) ),  X:[256,K], W:[K,512]
// ===========================================================================
__global__ __launch_bounds__(256) void global_proj_kernel(
    const float* __restrict__ X, const float* __restrict__ W,
    const float* __restrict__ b, const float* __restrict__ g,
    const float* __restrict__ bt, float* __restrict__ out, int K)
{
    __shared__ float sm[16 * LDST];
    const int tid = threadIdx.x, wave = tid >> 5, lane = tid & 31;
    const int half = lane >> 4, lmod = lane & 15;
    const int row0 = blockIdx.x * 16, colbase = wave * 64;

    v8f acc0 = ZERO8, acc1 = ZERO8, acc2 = ZERO8, acc3 = ZERO8;
    const float* xr = X + (size_t)(row0 + lmod) * K;
    for (int k0 = 0; k0 < K; k0 += 4) {
        const int ka = k0 + 2 * half;
        v2f a; a.x = xr[ka]; a.y = xr[ka + 1];
        const float* w0 = W + (size_t)ka * 512;
        v2f bb;
        bb.x = w0[colbase + lmod];      bb.y = w0[512 + colbase + lmod];
        acc0 = wmma_f32(a, bb, acc0);
        bb.x = w0[colbase + 16 + lmod]; bb.y = w0[512 + colbase + 16 + lmod];
        acc1 = wmma_f32(a, bb, acc1);
        bb.x = w0[colbase + 32 + lmod]; bb.y = w0[512 + colbase + 32 + lmod];
        acc2 = wmma_f32(a, bb, acc2);
        bb.x = w0[colbase + 48 + lmod]; bb.y = w0[512 + colbase + 48 + lmod];
        acc3 = wmma_f32(a, bb, acc3);
    }
    {
        v8f* accs[4] = {&acc0, &acc1, &acc2, &acc3};
#pragma unroll
        for (int t = 0; t < 4; ++t) {
            const int col = colbase + t * 16 + lmod;
            const float bias = b[col];
#pragma unroll
            for (int e = 0; e < 8; ++e)
                sm[(e + 8 * half) * LDST + col] = (*accs[t])[e] + bias;
        }
    }
    __syncthreads();

    for (int r = wave; r < 16; r += 8) {
        float s1 = 0.f, s2 = 0.f;
        for (int c = lane; c < 512; c += 32) {
            const float x = sm[r * LDST + c]; s1 += x; s2 += x * x;
        }
#pragma unroll
        for (int off = 16; off > 0; off >>= 1) {
            s1 += __shfl_xor(s1, off); s2 += __shfl_xor(s2, off);
        }
        const float mu   = s1 * (1.0f / 512.0f);
        const float rstd = rsqrtf(s2 * (1.0f / 512.0f) - mu * mu + 1e-5f);
        float ss = 0.f;
        for (int c = lane; c < 512; c += 32) {
            const float y = (sm[r * LDST + c] - mu) * rstd * g[c] + bt[c];
            sm[r * LDST + c] = y; ss += y * y;
        }
#pragma unroll
        for (int off = 16; off > 0; off >>= 1) ss += __shfl_xor(ss, off);
        const float inv = 1.0f / fmaxf(sqrtf(ss), 1e-12f);
        float* o = out + (size_t)(row0 + r) * 512;
        for (int c = lane; c < 512; c += 32) o[c] = sm[r * LDST + c] * inv;
    }
}

// ===========================================================================
// logits: S[256,256] = Vg @ Tg^T  (K=512).  Block = 16 rows, wave = 32 cols.
// ===========================================================================
__global__ __launch_bounds__(256) void logits_kernel(
    const float* __restrict__ V, const float* __restrict__ T,
    float* __restrict__ S)
{
    const int tid = threadIdx.x, wave = tid >> 5, lane = tid & 31;
    const int half = lane >> 4, lmod = lane & 15;
    const int row0 = blockIdx.x * 16, colbase = wave * 32;

    v8f acc0 = ZERO8, acc1 = ZERO8;
    const float* vr = V + (size_t)(row0 + lmod) * 512;
    const float* t0 = T + (size_t)(colbase + lmod) * 512;
    const float* t1 = T + (size_t)(colbase + 16 + lmod) * 512;
    for (int k0 = 0; k0 < 512; k0 += 4) {
        const int ka = k0 + 2 * half;
        v2f a;  a.x = vr[ka]; a.y = vr[ka + 1];
        v2f b0; b0.x = t0[ka]; b0.y = t0[ka + 1];
        v2f b1; b1.x = t1[ka]; b1.y = t1[ka + 1];
        acc0 = wmma_f32(a, b0, acc0);
        acc1 = wmma_f32(a, b1, acc1);
    }
#pragma unroll
    for (int e = 0; e < 8; ++e) {
        const int m = row0 + e + 8 * half;
        S[(size_t)m * 256 + colbase + lmod]      = acc0[e];
        S[(size_t)m * 256 + colbase + 16 + lmod] = acc1[e];
    }
}

// ===========================================================================
// Global CLIP loss from raw logits S.  One block of 256 threads.
// ===========================================================================
__global__ __launch_bounds__(256) void clip_loss_kernel(
    const float* __restrict__ S, const float* __restrict__ lsc,
    float* __restrict__ out0)
{
    __shared__ float red[256];
    const int i = threadIdx.x;
    const float scale = fminf(expf(lsc[0]), 100.0f);
    const float d = scale * S[(size_t)i * 256 + i];
    float mr = -3.4e38f, mc = -3.4e38f;
    for (int j = 0; j < 256; ++j) {
        mr = fmaxf(mr, scale * S[(size_t)i * 256 + j]);
        mc = fmaxf(mc, scale * S[(size_t)j * 256 + i]);
    }
    float sr = 0.f, sc = 0.f;
    for (int j = 0; j < 256; ++j) {
        sr += expf(scale * S[(size_t)i * 256 + j] - mr);
        sc += expf(scale * S[(size_t)j * 256 + i] - mc);
    }
    red[i] = (d - (mr + logf(sr))) + (d - (mc + logf(sc)));
    __syncthreads();
    for (int off = 128; off > 0; off >>= 1) {
        if (i < off) red[i] += red[i + off];
        __syncthreads();
    }
    if (i == 0) out0[0] = -red[0] / (2.0f * 256.0f);
}

// ===========================================================================
// sim[b] = v_proj[b] (64x512) @ t_proj[b]^T (512x77), raw.  1 block / batch.
// ===========================================================================
__global__ __launch_bounds__(256) void sim_kernel(
    const float* __restrict__ VP, const float* __restrict__ TP,
    float* __restrict__ SIM)
{
    const int b = blockIdx.x, tid = threadIdx.x;
    const int wave = tid >> 5, lane = tid & 31, half = lane >> 4, lmod = lane & 15;
    const float* Vb = VP + (size_t)b * 64 * 512;
    const float* Tb = TP + (size_t)b * 77 * 512;

    for (int T = wave; T < 20; T += 8) {       // 4 M-tiles x 5 N-tiles
        const int mt = T / 5, nt = T % 5;
        const int arow = mt * 16 + lmod;
        const int brow = nt * 16 + lmod;
        const int browc = brow < 77 ? brow : 0;          // clamped (safe) addr
        const float bm = brow < 77 ? 1.0f : 0.0f;        // zero mask
        const float* Ar = Vb + (size_t)arow * 512;
        const float* Br = Tb + (size_t)browc * 512;
        v8f c = ZERO8;
        for (int k0 = 0; k0 < 512; k0 += 4) {
            const int ka = k0 + 2 * half;
            v2f a;  a.x = Ar[ka];       a.y = Ar[ka + 1];
            v2f bb; bb.x = Br[ka] * bm; bb.y = Br[ka + 1] * bm;
            c = wmma_f32(a, bb, c);
        }
        if (brow < 77) {
#pragma unroll
            for (int e = 0; e < 8; ++e) {
                const int m = mt * 16 + e + 8 * half;
                SIM[((size_t)b * 64 + m) * 77 + brow] = c[e];
            }
        }
    }
}

// ===========================================================================
// Unit-level mutual-NN loss partials per batch.  160 threads (5 waves).
// ===========================================================================
__global__ __launch_bounds__(160) void unit_loss_kernel(
    const float* __restrict__ SIM, const float* __restrict__ lsc,
    float* __restrict__ lossP, float* __restrict__ mP)
{
    __shared__ float s[64 * 77];
    __shared__ float rowlse[64]; __shared__ int rowarg[64];
    __shared__ float collse[77]; __shared__ int colarg[77];
    __shared__ float redl[64], redm[64];
    const int b = blockIdx.x, tid = threadIdx.x;
    const float scale = fminf(expf(lsc[0]), 100.0f);
    for (int i = tid; i < 64 * 77; i += 160)
        s[i] = SIM[(size_t)b * 64 * 77 + i] * scale;
    __syncthreads();
    if (tid < 64) {
        const int i = tid;
        float mx = -3.4e38f; int arg = 0;
        for (int j = 0; j < 77; ++j) { const float v = s[i * 77 + j]; if (v > mx) { mx = v; arg = j; } }
        float se = 0.f;
        for (int j = 0; j < 77; ++j) se += expf(s[i * 77 + j] - mx);
        rowlse[i] = mx + logf(se); rowarg[i] = arg;
    } else if (tid < 141) {
        const int j = tid - 64;
        float mx = -3.4e38f; int arg = 0;
        for (int i = 0; i < 64; ++i) { const float v = s[i * 77 + j]; if (v > mx) { mx = v; arg = i; } }
        float se = 0.f;
        for (int i = 0; i < 64; ++i) se += expf(s[i * 77 + j] - mx);
        collse[j] = mx + logf(se); colarg[j] = arg;
    }
    __syncthreads();
    if (tid < 64) {
        const int i = tid, j = rowarg[i];
        const int mut = (colarg[j] == i) ? 1 : 0;
        const float lp1 = s[i * 77 + j] - rowlse[i];
        const float lp2 = s[i * 77 + j] - collse[j];
        redl[i] = mut ? -(lp1 + lp2) : 0.0f;
        redm[i] = (float)mut;
    }
    __syncthreads();
    if (tid == 0) {
        float a = 0.f, m = 0.f;
        for (int i = 0; i < 64; ++i) { a += redl[i]; m += redm[i]; }
        lossP[b] = a; mP[b] = m;
    }
}

// ===========================================================================
// Diversity partial per batch: sum_{i!=j} |u_i.u_j| / (||u_i|| ||u_j||).
// Gram matrix via WMMA; diagonal supplies the norms.
// ===========================================================================
__global__ __launch_bounds__(256) void diversity_kernel(
    const float* __restrict__ U, int n, float* __restrict__ part)
{
    __shared__ float G[80 * 80];
    __shared__ float nrm[80];
    __shared__ float red[8];
    const int b = blockIdx.x, tid = threadIdx.x;
    const int wave = tid >> 5, lane = tid & 31, half = lane >> 4, lmod = lane & 15;
    const int ntile = (n + 15) >> 4;
    const float* Ub = U + (size_t)b * n * 512;

    for (int T = wave; T < ntile * ntile; T += 8) {
        const int mt = T / ntile, nt = T % ntile;
        const int ar = mt * 16 + lmod, br = nt * 16 + lmod;
        const int arc = ar < n ? ar : 0, brc = br < n ? br : 0;
        const float am = ar < n ? 1.0f : 0.0f, bm2 = br < n ? 1.0f : 0.0f;
        const float* Ar = Ub + (size_t)arc * 512;
        const float* Br = Ub + (size_t)brc * 512;
        v8f c = ZERO8;
        for (int k0 = 0; k0 < 512; k0 += 4) {
            const int ka = k0 + 2 * half;
            v2f a;  a.x = Ar[ka] * am;   a.y = Ar[ka + 1] * am;
            v2f bb; bb.x = Br[ka] * bm2; bb.y = Br[ka + 1] * bm2;
            c = wmma_f32(a, bb, c);
        }
#pragma unroll
        for (int e = 0; e < 8; ++e)
            G[(mt * 16 + e + 8 * half) * 80 + nt * 16 + lmod] = c[e];
    }
    __syncthreads();
    if (tid < n) nrm[tid] = fmaxf(sqrtf(G[tid * 80 + tid]), 1e-12f);
    __syncthreads();
    float s = 0.f;
    for (int idx = tid; idx < n * n; idx += 256) {
        const int i = idx / n, j = idx - i * n;
        if (i != j) s += fabsf(G[i * 80 + j]) / (nrm[i] * nrm[j]);
    }
#pragma unroll
    for (int off = 16; off > 0; off >>= 1) s += __shfl_xor(s, off);
    if (lane == 0) red[wave] = s;
    __syncthreads();
    if (tid == 0) {
        float t = 0.f;
        for (int w = 0; w < 8; ++w) t += red[w];
        part[b] = t;
    }
}

// ===========================================================================
// Deterministic reduction of partials -> loss_unit, loss_div.
// P layout: [0..255]=div_v, [256..511]=div_t, [512..767]=unit loss, [768..1023]=unit m
// ===========================================================================
__global__ __launch_bounds__(256) void finalize_kernel(
    const float* __restrict__ P, float* __restrict__ out)
{
    __shared__ float r0[256], r1[256], r2[256], r3[256];
    const int t = threadIdx.x;
    r0[t] = P[t]; r1[t] = P[256 + t]; r2[t] = P[512 + t]; r3[t] = P[768 + t];
    __syncthreads();
    for (int off = 128; off > 0; off >>= 1) {
        if (t < off) { r0[t] += r0[t + off]; r1[t] += r1[t + off];
                       r2[t] += r2[t + off]; r3[t] += r3[t + off]; }
        __syncthreads();
    }
    if (t == 0) {
        out[1] = r2[0] / fmaxf(2.0f * r3[0], 1.0f);
        out[2] = 0.5f * (r0[0] / (256.0f * 64.0f * 63.0f) +
                         r1[0] / (256.0f * 77.0f * 76.0f));
    }
}

// ===========================================================================
extern "C" void kernel_launch(void* const* d_in, const int* in_sizes, int n_in,
                              void* d_out, int out_size, void* d_ws, size_t ws_size,
                              hipStream_t stream)
{
    const float* visual_units = (const float*)d_in[0];
    const float* text_units   = (const float*)d_in[1];
    const float* visual_cls   = (const float*)d_in[2];
    const float* text_cls     = (const float*)d_in[3];
    const float* vW1 = (const float*)d_in[4];  const float* vb1 = (const float*)d_in[5];
    const float* vg  = (const float*)d_in[6];  const float* vbt = (const float*)d_in[7];
    const float* vW2 = (const float*)d_in[8];  const float* vb2 = (const float*)d_in[9];
    const float* tW1 = (const float*)d_in[10]; const float* tb1 = (const float*)d_in[11];
    const float* tg  = (const float*)d_in[12]; const float* tbt = (const float*)d_in[13];
    const float* tW2 = (const float*)d_in[14]; const float* tb2 = (const float*)d_in[15];
    const float* gvW = (const float*)d_in[16]; const float* gvb = (const float*)d_in[17];
    const float* gvg = (const float*)d_in[18]; const float* gvbt = (const float*)d_in[19];
    const float* gtW = (const float*)d_in[20]; const float* gtb = (const float*)d_in[21];
    const float* gtg = (const float*)d_in[22]; const float* gtbt = (const float*)d_in[23];
    const float* logit_scale = (const float*)d_in[24];

    float* out = (float*)d_out;
    float* ws  = (float*)d_ws;

    // output layout (flat concat, reference return order)
    constexpr size_t NVP = 256ull * 64 * 512;
    constexpr size_t NTP = 256ull * 77 * 512;
    constexpr size_t O_VPROJ = 3;
    constexpr size_t O_TPROJ = O_VPROJ + NVP;
    constexpr size_t O_VG    = O_TPROJ + NTP;
    constexpr size_t O_TG    = O_VG + 256ull * 512;
    constexpr size_t O_SIM   = O_TG + 256ull * 512;

    float* partials = ws;            // 1024 floats of partial sums
    float* S        = ws + 1024;     // 256x256 logits

    // unit MLP projections (WMMA fp32, fused LN/GELU/L2)
    unit_mlp_kernel<<<1024, 256, 0, stream>>>(visual_units, vW1, vb1, vg, vbt,
                                              vW2, vb2, out + O_VPROJ, 16384);
    unit_mlp_kernel<<<1232, 256, 0, stream>>>(text_units, tW1, tb1, tg, tbt,
                                              tW2, tb2, out + O_TPROJ, 19712);
    // global projections
    global_proj_kernel<<<16, 256, 0, stream>>>(visual_cls, gvW, gvb, gvg, gvbt,
                                               out + O_VG, 768);
    global_proj_kernel<<<16, 256, 0, stream>>>(text_cls, gtW, gtb, gtg, gtbt,
                                               out + O_TG, 512);
    // global CLIP loss
    logits_kernel<<<16, 256, 0, stream>>>(out + O_VG, out + O_TG, S);
    clip_loss_kernel<<<1, 256, 0, stream>>>(S, logit_scale, out);
    // unit-level sim + mutual-NN loss
    sim_kernel<<<256, 256, 0, stream>>>(out + O_VPROJ, out + O_TPROJ, out + O_SIM);
    unit_loss_kernel<<<256, 160, 0, stream>>>(out + O_SIM, logit_scale,
                                              partials + 512, partials + 768);
    // diversity loss (Gram via WMMA)
    diversity_kernel<<<256, 256, 0, stream>>>(visual_units, 64, partials);
    diversity_kernel<<<256, 256, 0, stream>>>(text_units, 77, partials + 256);
    // scalar losses
    finalize_kernel<<<1, 256, 0, stream>>>(partials, out);
}